// OptimizedTemporalGraphConvLayer_3521873182772
// MI455X (gfx1250) — compile-verified
//
#include <hip/hip_runtime.h>
#include <hip/hip_bf16.h>
#include <math.h>

// ---------------------------------------------------------------------------
// Shapes (compile-time constants from the reference)
// ---------------------------------------------------------------------------
#define Bb   8
#define Ss   256
#define Nn   512
#define Ii   32
#define HIDh 64
#define Oo   32
#define EPSf 1e-5f

typedef __attribute__((ext_vector_type(2))) float v2f;
typedef __attribute__((ext_vector_type(8))) float v8f;
typedef __attribute__((ext_vector_type(4))) unsigned int v4u;
typedef __attribute__((ext_vector_type(8))) int v8i;
typedef __attribute__((ext_vector_type(4))) int v4i;

// Native fp32 WMMA: D(16x16,f32) = A(16x4,f32) x B(4x16,f32) + C
__device__ __forceinline__ v8f wmma_f32(v2f a, v2f b, v8f c) {
  return __builtin_amdgcn_wmma_f32_16x16x4_f32(
      false, a, false, b, (short)0, c, false, false);
}

__device__ __forceinline__ float sigmoidf_(float x) {
  return 1.0f / (1.0f + __expf(-x));
}

__device__ __forceinline__ unsigned lds_off32(const void* p) {
  return (unsigned)(unsigned long long)(uintptr_t)p;  // low 32 bits = LDS offset
}

// ---------------------------------------------------------------------------
// Tensor Data Mover: async global->LDS DMA (tracked by TENSORcnt).
// 2D tile: `rows` lines of `cols` f32 elems, source line stride `stride_elems`.
// Optional LDS padding: insert (pad_amt+1) dwords every 2^(pad_iv+1) dwords,
// used to land rows at odd LDS strides (bank-conflict-free) for free.
// ---------------------------------------------------------------------------
__device__ __forceinline__ void tdm_load(unsigned lds_off, const void* gsrc,
                                         unsigned rows, unsigned cols,
                                         unsigned stride_elems,
                                         int pad_en, unsigned pad_iv,
                                         unsigned pad_amt)
{
  const unsigned long long ga = (unsigned long long)(uintptr_t)gsrc;
  const unsigned d0 = stride_elems;  // tensor_dim0 (covers tile cols)
  const unsigned d1 = rows;          // tensor_dim1
  const unsigned w0 = 1u;            // count=1 (valid, user mode)
  const unsigned w1 = lds_off;       // lds_addr
  const unsigned w2 = (unsigned)ga;  // global_addr[31:0]
  const unsigned w3 = (unsigned)((ga >> 32) & 0x01FFFFFFull) | (2u << 30); // type=2
  const unsigned u0 = (2u << 16) |   // data_size = 4B, wg_mask = 0
      (pad_en ? ((1u << 20) | (pad_iv << 22) | (pad_amt << 25)) : 0u);
  const unsigned u1 = (d0 & 0xFFFFu) << 16;                              // dim0 lo
  const unsigned u2 = ((d0 >> 16) & 0xFFFFu) | ((d1 & 0xFFFFu) << 16);   // dim0 hi | dim1 lo
  const unsigned u3 = ((d1 >> 16) & 0xFFFFu) | ((cols & 0xFFFFu) << 16); // dim1 hi | tile_dim0
  const unsigned u4 = (rows > 1u) ? (rows & 0xFFFFu) : 0u;               // tile_dim1
  const unsigned u5 = stride_elems;                                      // dim0_stride lo
  v4u g0 = { (unsigned)__builtin_amdgcn_readfirstlane((int)w0),
             (unsigned)__builtin_amdgcn_readfirstlane((int)w1),
             (unsigned)__builtin_amdgcn_readfirstlane((int)w2),
             (unsigned)__builtin_amdgcn_readfirstlane((int)w3) };
  v8i g1 = { __builtin_amdgcn_readfirstlane((int)u0),
             __builtin_amdgcn_readfirstlane((int)u1),
             __builtin_amdgcn_readfirstlane((int)u2),
             __builtin_amdgcn_readfirstlane((int)u3),
             __builtin_amdgcn_readfirstlane((int)u4),
             __builtin_amdgcn_readfirstlane((int)u5),
             0, 0 };
  v4i z4 = { 0, 0, 0, 0 };
  v8i z8 = { 0, 0, 0, 0, 0, 0, 0, 0 };
  __builtin_amdgcn_tensor_load_to_lds(g0, g1, z4, z4, z8, 0);
}

// ---------------------------------------------------------------------------
// Kernel 1: sp = relu(LN(xr@W1+b1)) @ W2 + b2 + xr        [2048, 512, 32]
// One block per (b,s); 4 waves; each wave owns 16-node row tiles.
// X tiles arrive via per-wave double-buffered TDM (padded to stride 33).
// ---------------------------------------------------------------------------
__global__ __launch_bounds__(128) void k_gcn(
    const float* __restrict__ x,
    const float* __restrict__ W1, const float* __restrict__ b1,
    const float* __restrict__ g1, const float* __restrict__ be1,
    const float* __restrict__ W2, const float* __restrict__ b2,
    float* __restrict__ spws)
{
  __shared__ float W1s[32][64];      // [k][h]
  __shared__ float W2s[64][32];      // [k][o]
  __shared__ float b1s[64], g1s[64], be1s[64], b2s[32];
  __shared__ __attribute__((aligned(16))) float Xt[4][2][16 * 33 + 4];
  __shared__ float Ht[4][16][66];    // per-wave hidden tile (padded)
  __shared__ float mus[4][16], rss[4][16];

  const int tid  = threadIdx.x;
  const int wave = tid >> 5;
  const int lane = tid & 31;
  const int bs   = blockIdx.x;

  for (int i = tid; i < 32 * 64; i += 128) W1s[i >> 6][i & 63] = W1[i];
  for (int i = tid; i < 64 * 32; i += 128) W2s[i >> 5][i & 31] = W2[i];
  if (tid < 64) { b1s[tid] = b1[tid]; g1s[tid] = g1[tid]; be1s[tid] = be1[tid]; }
  if (tid < 32) { b2s[tid] = b2[tid]; }

  const float* xb  = x    + (size_t)bs * Nn * Ii;
  float*       spb = spws + (size_t)bs * Nn * Oo;

  // pre-issue first X tile (flat 512 f32, pad 1 dword per 32 -> LDS stride 33)
  tdm_load(lds_off32(&Xt[wave][0][0]), xb + (size_t)(wave * 16) * Ii,
           1, 512, 512, 1, 4, 0);
  __syncthreads();

  const int m  = lane & 15;
  const int kh = lane >> 4;

  for (int i = 0; i < 8; ++i) {
    const int t   = wave + 4 * i;
    const int n0  = t * 16;
    const int buf = i & 1;

    if (i + 1 < 8) {
      const int tn = wave + 4 * (i + 1);
      tdm_load(lds_off32(&Xt[wave][buf ^ 1][0]), xb + (size_t)(tn * 16) * Ii,
               1, 512, 512, 1, 4, 0);
      __builtin_amdgcn_s_wait_tensorcnt(1);
    } else {
      __builtin_amdgcn_s_wait_tensorcnt(0);
    }
    const float* Xw = &Xt[wave][buf][0];

    // GEMM1: [16x32] @ [32x64], +b1 on store
#pragma unroll
    for (int ct = 0; ct < 4; ++ct) {
      v8f acc = {};
#pragma unroll
      for (int kk = 0; kk < 8; ++kk) {
        const int k0 = kk * 4 + kh * 2;
        v2f a, bfr;
        a.x   = Xw[m * 33 + k0];       a.y   = Xw[m * 33 + k0 + 1];
        bfr.x = W1s[k0][ct * 16 + m];  bfr.y = W1s[k0 + 1][ct * 16 + m];
        acc = wmma_f32(a, bfr, acc);
      }
      const int col = ct * 16 + m;
      const float bb = b1s[col];
#pragma unroll
      for (int v = 0; v < 8; ++v) {
        const int row = v + kh * 8;
        Ht[wave][row][col] = acc[v] + bb;
      }
    }
    __syncthreads();

    // LayerNorm over HID=64 per row, then ReLU
    if (lane < 16) {
      float s = 0.f;
#pragma unroll
      for (int c = 0; c < 64; ++c) s += Ht[wave][lane][c];
      const float mu = s * (1.0f / 64.0f);
      float vs = 0.f;
#pragma unroll
      for (int c = 0; c < 64; ++c) { float d = Ht[wave][lane][c] - mu; vs += d * d; }
      mus[wave][lane] = mu;
      rss[wave][lane] = rsqrtf(vs * (1.0f / 64.0f) + EPSf);
    }
    __syncthreads();
    {
      const int row = lane >> 1, c0 = (lane & 1) * 32;
      const float mu = mus[wave][row], rs = rss[wave][row];
#pragma unroll
      for (int j = 0; j < 32; ++j) {
        const int c = c0 + j;
        float hv = (Ht[wave][row][c] - mu) * rs * g1s[c] + be1s[c];
        Ht[wave][row][c] = fmaxf(hv, 0.0f);
      }
    }
    __syncthreads();

    // GEMM2: [16x64] @ [64x32] + b2 + residual(x)
#pragma unroll
    for (int ct = 0; ct < 2; ++ct) {
      const int col = ct * 16 + m;
      const float bv = b2s[col];
      v8f acc = { bv, bv, bv, bv, bv, bv, bv, bv };
#pragma unroll
      for (int kk = 0; kk < 16; ++kk) {
        const int k0 = kk * 4 + kh * 2;
        v2f a, bfr;
        a.x   = Ht[wave][m][k0];  a.y   = Ht[wave][m][k0 + 1];
        bfr.x = W2s[k0][col];     bfr.y = W2s[k0 + 1][col];
        acc = wmma_f32(a, bfr, acc);
      }
#pragma unroll
      for (int v = 0; v < 8; ++v) {
        const int row = v + kh * 8;
        spb[(n0 + row) * Oo + col] = acc[v] + Xw[row * 33 + col];
      }
    }
    __syncthreads();
  }
}

// ---------------------------------------------------------------------------
// Kernel 2: seq = LN2(tanh(adj @ sp_b)), scattered to [B*N, S, O] layout.
// One block per (bs, 1/8 of nodes); 4 waves, each owns one 16-node tile.
// sp_b (512x32) streamed through LDS in double-buffered TDM K-panels;
// adj rows come straight from L2 (1 MB, 2048x reuse) with prefetch.
// ---------------------------------------------------------------------------
__global__ __launch_bounds__(128) void k_adj(
    const float* __restrict__ spws, const float* __restrict__ adj,
    const float* __restrict__ g2,   const float* __restrict__ be2,
    float* __restrict__ seqws)
{
  __shared__ __attribute__((aligned(16))) float SP[2][128 * 32];
  __shared__ float Tt[4][16][33];    // per-wave LN staging tile
  __shared__ float mus[4][16], rss[4][16];
  __shared__ float g2s[32], be2s[32];

  const int tid  = threadIdx.x;
  const int wave = tid >> 5;
  const int lane = tid & 31;
  const int bid  = blockIdx.x;
  const int bs   = bid >> 3;
  const int seg  = bid & 7;
  const int b    = bs >> 8;
  const int s    = bs & 255;
  const int n0   = (seg * 4 + wave) * 16;

  if (tid < 32) { g2s[tid] = g2[tid]; be2s[tid] = be2[tid]; }

  const int m  = lane & 15;
  const int kh = lane >> 4;

  const float* spsrc = spws + (size_t)bs * Nn * Oo;
  if (wave == 0)
    tdm_load(lds_off32(&SP[0][0]), spsrc, 1, 4096, 4096, 0, 0, 0);

  v8f acc0 = {}, acc1 = {};
  const float* adjr = adj + (size_t)(n0 + m) * Nn;

  for (int p = 0; p < 4; ++p) {
    if (wave == 0) {
      if (p < 3) {
        tdm_load(lds_off32(&SP[(p + 1) & 1][0]), spsrc + (p + 1) * 4096,
                 1, 4096, 4096, 0, 0, 0);
        __builtin_amdgcn_s_wait_tensorcnt(1);
      } else {
        __builtin_amdgcn_s_wait_tensorcnt(0);
      }
    }
    __syncthreads();                       // panel p ready for all waves
    const float* SPb = &SP[p & 1][0];

    if (p < 3)
      __builtin_prefetch(adjr + (p + 1) * 128 + kh * 64, 0, 0);

#pragma unroll 8
    for (int kk = 0; kk < 32; ++kk) {
      const int kl = kk * 4 + kh * 2;      // local K in panel
      const int kg = p * 128 + kl;         // global K
      v2f a, b0, b1v;
      const float2 av = *(const float2*)(adjr + kg);
      a.x = av.x; a.y = av.y;
      b0.x  = SPb[kl * 32 + m];        b0.y  = SPb[(kl + 1) * 32 + m];
      b1v.x = SPb[kl * 32 + 16 + m];   b1v.y = SPb[(kl + 1) * 32 + 16 + m];
      acc0 = wmma_f32(a, b0, acc0);
      acc1 = wmma_f32(a, b1v, acc1);
    }
    __syncthreads();                       // all done reading before overwrite
  }

  // tanh, stage tile for per-row LayerNorm over O=32
#pragma unroll
  for (int v = 0; v < 8; ++v) {
    const int row = v + kh * 8;
    Tt[wave][row][m]      = tanhf(acc0[v]);
    Tt[wave][row][16 + m] = tanhf(acc1[v]);
  }
  __syncthreads();

  if (lane < 16) {
    float sm = 0.f;
#pragma unroll
    for (int c = 0; c < 32; ++c) sm += Tt[wave][lane][c];
    const float mu = sm * (1.0f / 32.0f);
    float vs = 0.f;
#pragma unroll
    for (int c = 0; c < 32; ++c) { float d = Tt[wave][lane][c] - mu; vs += d * d; }
    mus[wave][lane] = mu;
    rss[wave][lane] = rsqrtf(vs * (1.0f / 32.0f) + EPSf);
  }
  __syncthreads();

  // scatter to seq layout [(b*N+n)][s][f]
  float* outb = seqws + (((size_t)(b * Nn + n0)) * Ss + s) * Oo;
#pragma unroll
  for (int j = 0; j < 16; ++j) {
    const float mu = mus[wave][j], rs = rss[wave][j];
    const float vv = (Tt[wave][j][lane] - mu) * rs * g2s[lane] + be2s[lane];
    outb[(size_t)j * Ss * Oo + lane] = vv;
  }
}

// ---------------------------------------------------------------------------
// Kernel 3: fused GRU — gi GEMM + gh GEMM + gates, 256 sequential steps.
// 256 persistent single-wave blocks; each owns 16 rows (bn) of hidden state.
// Weights (untransposed, padded stride 34 -> contiguous b64 B-frags) and the
// per-step 16x32 seq tiles arrive via TDM; seq is double-buffered so the DMA
// for step s+1 overlaps the 96 WMMAs of step s.
// ---------------------------------------------------------------------------
__global__ __launch_bounds__(32) void k_gru(
    const float* __restrict__ seqws,
    const float* __restrict__ w_ih, const float* __restrict__ w_hh,
    const float* __restrict__ b_ih, const float* __restrict__ b_hh,
    float* __restrict__ out)
{
  __shared__ __attribute__((aligned(16))) float Wi[96 * 34 + 4];  // [j][k] stride 34
  __shared__ __attribute__((aligned(16))) float Wh[96 * 34 + 4];
  __shared__ __attribute__((aligned(16))) float Seq[2][16 * 34 + 4];
  __shared__ float Htile[16][33];

  const int lane = threadIdx.x & 31;
  const int m  = lane & 15;
  const int kh = lane >> 4;
  const size_t bn0 = (size_t)blockIdx.x * 16;

  const float* seqb0 = seqws + bn0 * (size_t)(Ss * Oo);

  // weights: flat 3072 f32, pad 2 dwords per 32 -> stride 34
  tdm_load(lds_off32(&Wi[0]), w_ih, 1, 3072, 3072, 1, 4, 1);
  tdm_load(lds_off32(&Wh[0]), w_hh, 1, 3072, 3072, 1, 4, 1);
  // first seq tile: 16 rows x 32 cols, source row stride S*O, stride 34 in LDS
  tdm_load(lds_off32(&Seq[0][0]), seqb0, 16, 32, Ss * Oo, 1, 4, 1);

  float bi[6], bh[6];
#pragma unroll
  for (int ct = 0; ct < 6; ++ct) {
    bi[ct] = b_ih[ct * 16 + m];
    bh[ct] = b_hh[ct * 16 + m];
  }
#pragma unroll
  for (int j = 0; j < 16; ++j) Htile[j][lane] = 0.0f;
  v8f h0v = {}, h1v = {};
  __syncthreads();

  const int bo = (int)(bn0 >> 9);        // batch index
  const int no = (int)(bn0 & 511);       // node index

  for (int st = 0; st < Ss; ++st) {
    // overlap: DMA step st+1 while computing step st
    if (st + 1 < Ss) {
      tdm_load(lds_off32(&Seq[(st + 1) & 1][0]), seqb0 + (size_t)(st + 1) * Oo,
               16, 32, Ss * Oo, 1, 4, 1);
      __builtin_amdgcn_s_wait_tensorcnt(1);
    } else {
      __builtin_amdgcn_s_wait_tensorcnt(0);
    }
    const float* Sq = &Seq[st & 1][0];

    // A frags: seq rows (gi GEMM) and hprev rows (gh GEMM)
    v2f as[8], ah[8];
#pragma unroll
    for (int kk = 0; kk < 8; ++kk) {
      const int k0 = kk * 4 + kh * 2;
      as[kk].x = Sq[m * 34 + k0];  as[kk].y = Sq[m * 34 + k0 + 1];
      ah[kk].x = Htile[m][k0];     ah[kk].y = Htile[m][k0 + 1];
    }

    v8f gi[6], gh[6];
#pragma unroll
    for (int ct = 0; ct < 6; ++ct) {
      const int col = ct * 16 + m;
      const float bvi = bi[ct], bvh = bh[ct];
      v8f ai  = { bvi, bvi, bvi, bvi, bvi, bvi, bvi, bvi };
      v8f ah8 = { bvh, bvh, bvh, bvh, bvh, bvh, bvh, bvh };
#pragma unroll
      for (int kk = 0; kk < 8; ++kk) {
        const int k0 = kk * 4 + kh * 2;
        v2f bwi, bwh;
        bwi.x = Wi[col * 34 + k0];  bwi.y = Wi[col * 34 + k0 + 1];
        bwh.x = Wh[col * 34 + k0];  bwh.y = Wh[col * 34 + k0 + 1];
        ai  = wmma_f32(as[kk], bwi, ai);
        ah8 = wmma_f32(ah[kk], bwh, ah8);
      }
      gi[ct] = ai;
      gh[ct] = ah8;
    }

    // gates: r=sig(0,1) z=sig(2,3) n=tanh(4,5 with r*gh)
    v8f hn0, hn1;
#pragma unroll
    for (int half = 0; half < 2; ++half) {
      v8f hp = half ? h1v : h0v;
      v8f hn;
#pragma unroll
      for (int v = 0; v < 8; ++v) {
        const float r  = sigmoidf_(gi[half][v]     + gh[half][v]);
        const float z  = sigmoidf_(gi[2 + half][v] + gh[2 + half][v]);
        const float nn = tanhf(gi[4 + half][v] + r * gh[4 + half][v]);
        hn[v] = (1.0f - z) * nn + z * hp[v];
      }
      if (half) hn1 = hn; else hn0 = hn;
    }
    h0v = hn0; h1v = hn1;

    // store to out[b][s][n][f] and refresh LDS mirror of h
    float* ob = out + (((size_t)bo * Ss + st) * Nn + no) * Oo;
#pragma unroll
    for (int v = 0; v < 8; ++v) {
      const int row = v + kh * 8;
      ob[(size_t)row * Oo + m]      = hn0[v];
      ob[(size_t)row * Oo + 16 + m] = hn1[v];
    }
    __syncthreads();
#pragma unroll
    for (int v = 0; v < 8; ++v) {
      const int row = v + kh * 8;
      Htile[row][m]      = hn0[v];
      Htile[row][16 + m] = hn1[v];
    }
    __syncthreads();
  }
}

// ---------------------------------------------------------------------------
extern "C" void kernel_launch(void* const* d_in, const int* in_sizes, int n_in,
                              void* d_out, int out_size, void* d_ws, size_t ws_size,
                              hipStream_t stream) {
  (void)in_sizes; (void)n_in; (void)out_size; (void)ws_size;
  const float* x    = (const float*)d_in[0];
  const float* adj  = (const float*)d_in[1];
  const float* W1   = (const float*)d_in[2];
  const float* b1   = (const float*)d_in[3];
  const float* g1   = (const float*)d_in[4];
  const float* be1  = (const float*)d_in[5];
  const float* W2   = (const float*)d_in[6];
  const float* b2   = (const float*)d_in[7];
  const float* g2   = (const float*)d_in[8];
  const float* be2  = (const float*)d_in[9];
  const float* w_ih = (const float*)d_in[10];
  const float* w_hh = (const float*)d_in[11];
  const float* b_ih = (const float*)d_in[12];
  const float* b_hh = (const float*)d_in[13];
  float* out = (float*)d_out;

  float* spws  = (float*)d_ws;                              // 2048*512*32 f32
  float* seqws = spws + (size_t)Bb * Ss * Nn * Oo;          // 4096*256*32 f32

  k_gcn<<<Bb * Ss, 128, 0, stream>>>(x, W1, b1, g1, be1, W2, b2, spws);
  k_adj<<<Bb * Ss * 8, 128, 0, stream>>>(spws, adj, g2, be2, seqws);
  k_gru<<<(Bb * Nn) / 16, 32, 0, stream>>>(seqws, w_ih, w_hh, b_ih, b_hh, out);
}